// LIFLayer_6107443495397
// MI455X (gfx1250) — compile-verified
//
#include <hip/hip_runtime.h>
#include <hip/hip_bf16.h>

// CDNA5 / gfx1250: wave32, WMMA 16x16 shapes.
typedef __attribute__((ext_vector_type(2))) float v2f;
typedef __attribute__((ext_vector_type(8))) float v8f;

__device__ __forceinline__ v8f wmma_f32(v2f a, v2f b, v8f c) {
  // 8 args: (neg_a, A, neg_b, B, c_mod, C, reuse_a, reuse_b)
  return __builtin_amdgcn_wmma_f32_16x16x4_f32(
      false, a, false, b, (short)0, c, false, false);
}

// -----------------------------------------------------------------------------
// GEMM: h[M x N] = x[M x K] * W[N x K]^T + bias[N]
// M = T*B = 32768, N = H = 512, K = I = 512.
// Register-blocked: each wave computes a 32x32 output (2x2 tiles of 16x16),
// so each A/B fragment is reused by two WMMAs (1 load : 1 wmma).
// 8 waves per 256-thread block. No divergence -> EXEC all-ones (WMMA req).
//
// VGPR layouts (ISA 7.12.2), 32-bit element WMMA:
//   A 16x4 f32 : lane L<16 -> row M=L, {K=0,K=1}; lane L>=16 -> row M=L-16, {K=2,K=3}
//   B 4x16 f32 : lane L<16 -> col N=L, {K=0,K=1}; lane L>=16 -> col N=L-16, {K=2,K=3}
//   C/D 16x16  : VGPR r -> M = r + 8*(lane>=16), N = lane&15
// Per-lane A and B data are K-contiguous float2 in row-major x / row-major W.
// -----------------------------------------------------------------------------
template <int KS>  // KS > 0: compile-time K (removes remainder path); KS==0: runtime
__global__ __launch_bounds__(256) void lif_gemm_wmma_f32_2x2(
    const float* __restrict__ x,     // [M, K]
    const float* __restrict__ W,     // [N, K]
    const float* __restrict__ bias,  // [N]
    float* __restrict__ h,           // [M, N]
    int M, int N, int K_rt) {
  const int K = (KS > 0) ? KS : K_rt;

  const int lane = threadIdx.x & 31;
  const int wave = threadIdx.x >> 5;

  const int tiles_n = N >> 5;                       // 32-wide tiles along N
  const int tile    = blockIdx.x * 8 + wave;        // global wave-tile id
  const int tm      = (tile / tiles_n) << 5;        // tile row (M), 32-high
  const int tn      = (tile % tiles_n) << 5;        // tile col (N), 32-wide

  const int l15   = lane & 15;
  const int khalf = (lane >> 4) << 1;               // 0 (lanes 0-15) / 2 (16-31)

  const float* __restrict__ ap0 = x + (size_t)(tm + l15) * K + khalf;
  const float* __restrict__ ap1 = ap0 + (size_t)16 * K;
  const float* __restrict__ bp0 = W + (size_t)(tn + l15) * K + khalf;
  const float* __restrict__ bp1 = bp0 + (size_t)16 * K;

  v8f c00 = {0.f,0.f,0.f,0.f,0.f,0.f,0.f,0.f};
  v8f c01 = c00, c10 = c00, c11 = c00;

#pragma unroll 4
  for (int k = 0; k < K; k += 4) {
    const v2f a0 = *(const v2f*)(ap0 + k);
    const v2f a1 = *(const v2f*)(ap1 + k);
    const v2f b0 = *(const v2f*)(bp0 + k);
    const v2f b1 = *(const v2f*)(bp1 + k);
    c00 = wmma_f32(a0, b0, c00);
    c01 = wmma_f32(a0, b1, c01);
    c10 = wmma_f32(a1, b0, c10);
    c11 = wmma_f32(a1, b1, c11);
  }

  const int n0 = tn + l15;
  const int n1 = n0 + 16;
  const int mb = tm + ((lane >> 4) << 3);           // +8 for upper half-wave
  const float bn0 = bias[n0];
  const float bn1 = bias[n1];

#pragma unroll
  for (int r = 0; r < 8; ++r) {
    h[(size_t)(mb + r) * N + n0]      = c00[r] + bn0;
    h[(size_t)(mb + r) * N + n1]      = c01[r] + bn1;
    h[(size_t)(mb + 16 + r) * N + n0] = c10[r] + bn0;
    h[(size_t)(mb + 16 + r) * N + n1] = c11[r] + bn1;
  }
}

// Fallback: one 16x16 tile per wave (handles dims not divisible by 32).
__global__ __launch_bounds__(256) void lif_gemm_wmma_f32_1x1(
    const float* __restrict__ x, const float* __restrict__ W,
    const float* __restrict__ bias, float* __restrict__ h,
    int M, int N, int K) {
  const int lane = threadIdx.x & 31;
  const int wave = threadIdx.x >> 5;
  const int tiles_n = N >> 4;
  const int tile    = blockIdx.x * 8 + wave;
  const int tm      = (tile / tiles_n) << 4;
  const int tn      = (tile % tiles_n) << 4;
  const int l15   = lane & 15;
  const int khalf = (lane >> 4) << 1;
  const float* ap = x + (size_t)(tm + l15) * K + khalf;
  const float* bp = W + (size_t)(tn + l15) * K + khalf;
  v8f c = {0.f,0.f,0.f,0.f,0.f,0.f,0.f,0.f};
  for (int k = 0; k < K; k += 4) {
    v2f a  = *(const v2f*)(ap + k);
    v2f bm = *(const v2f*)(bp + k);
    c = wmma_f32(a, bm, c);
  }
  const int n = tn + l15;
  const int mb = tm + ((lane >> 4) << 3);
  const float bn = bias[n];
#pragma unroll
  for (int r = 0; r < 8; ++r) h[(size_t)(mb + r) * N + n] = c[r] + bn;
}

// -----------------------------------------------------------------------------
// LIF scan over T. TAU=2, V_TH=1, V_RESET=0, decay_input=False, hard reset.
//   v <- v - v/TAU + h_t = 0.5*v + h_t ; s = (v >= 1) ; v <- v*(1-s)
// One thread per (b, h) neuron; plane-contiguous (coalesced) access per t.
// h typically still resident in the 192MB L2 from the GEMM pass.
// -----------------------------------------------------------------------------
__global__ __launch_bounds__(256) void lif_scan(
    const float* __restrict__ h,   // [T, plane]
    float* __restrict__ out,       // [T, plane]
    int T, int plane) {
  const int idx = blockIdx.x * blockDim.x + threadIdx.x;
  if (idx >= plane) return;

  float v = 0.f;
  for (int t = 0; t < T; ++t) {
    const float ht = h[(size_t)t * plane + idx];
    v = __builtin_fmaf(v, 0.5f, ht);
    const float s = (v >= 1.0f) ? 1.0f : 0.0f;
    v = (s != 0.0f) ? 0.0f : v;
    out[(size_t)t * plane + idx] = s;
  }
}

extern "C" void kernel_launch(void* const* d_in, const int* in_sizes, int n_in,
                              void* d_out, int out_size, void* d_ws, size_t ws_size,
                              hipStream_t stream) {
  const float* x    = (const float*)d_in[0];  // (T, B, I) f32
  const float* W    = (const float*)d_in[1];  // (H, I)    f32
  const float* bias = (const float*)d_in[2];  // (H,)      f32
  float* out        = (float*)d_out;          // (T, B, H) f32
  float* h_ws       = (float*)d_ws;           // scratch: M*N f32 = 64 MB

  const int H = in_sizes[2];                  // 512
  const int I = in_sizes[1] / H;              // 512
  const int M = in_sizes[0] / I;              // T*B = 32768
  const int T = 256;                          // reference setup is fixed
  const int plane = (M / T) * H;              // B*H = 65536

  if ((M % 32) == 0 && (H % 32) == 0 && (I % 4) == 0) {
    const int wave_tiles = (M >> 5) * (H >> 5);   // 16384
    if (I == 512) {
      lif_gemm_wmma_f32_2x2<512><<<wave_tiles / 8, 256, 0, stream>>>(
          x, W, bias, h_ws, M, H, I);
    } else {
      lif_gemm_wmma_f32_2x2<0><<<wave_tiles / 8, 256, 0, stream>>>(
          x, W, bias, h_ws, M, H, I);
    }
  } else {
    const int tiles = (M >> 4) * (H >> 4);
    lif_gemm_wmma_f32_1x1<<<tiles / 8, 256, 0, stream>>>(x, W, bias, h_ws, M, H, I);
  }

  // Sequential LIF dynamics over T, parallel across neurons.
  lif_scan<<<(plane + 255) / 256, 256, 0, stream>>>(h_ws, out, T, plane);
}